// MultiScaleHeatmapGenerator_57458072486269
// MI455X (gfx1250) — compile-verified
//
#include <hip/hip_runtime.h>
#include <hip/hip_bf16.h>

// MultiScaleHeatmapGenerator, MI455X (gfx1250).
//
// Roofline: output is 32*15*256*256 fp32 = 125.8 MB of pure stores; compute is
// ~0.9M exp() calls total (480 keypoints x 625 window pixels x <=3 scales).
// => strictly HBM-store-bound, floor ~5.4us at 23.3 TB/s. No contraction
// exists (scale combine is a max), so WMMA does not apply; the CDNA5 feature
// that matters is the store temporal-hint path: the output is write-once, so
// we stream it with non-temporal global_store_b128 (TH_STORE_NT) to avoid
// polluting/rinsing L2.
//
// Layout: one thread == one 16B quad (4 pixels of one row of one (b,k)
// channel). Block = 256 threads = 4 rows of 256 cols. Grid = B*K*(H/4).
// Keypoint is block-uniform -> scalar loads; off-window lanes store zero.

typedef __attribute__((ext_vector_type(4))) float v4f;

namespace {
constexpr int Bn = 32;
constexpr int Kn = 15;
constexpr int Hn = 256;
constexpr int Wn = 256;
constexpr int ROWS_PER_BLOCK  = 4;                    // 256 thr * 4 f32 = 4 rows
constexpr int BLOCKS_PER_CHAN = Hn / ROWS_PER_BLOCK;  // 64
} // namespace

__global__ __launch_bounds__(256)
void MultiScaleHeatmapGenerator_57458072486269_kernel(
    const int*   __restrict__ kp,   // [B,K,2] (x,y)
    const float* __restrict__ sw,   // [3]
    float*       __restrict__ out)  // [B,K,H,W]
{
    const int blk    = blockIdx.x;
    const int rowBlk = blk & (BLOCKS_PER_CHAN - 1);
    const int bk     = blk >> 6;                   // (b*K + k), BLOCKS_PER_CHAN==64

    // Block-uniform keypoint -> scalar loads.
    const int x = kp[2 * bk + 0];
    const int y = kp[2 * bk + 1];

    const int t   = threadIdx.x;
    const int row = rowBlk * ROWS_PER_BLOCK + (t >> 6);  // 64 threads per row
    const int col = (t & 63) << 2;                        // 4 pixels per thread

    float* p = out + (size_t)bk * (Hn * Wn) + (size_t)row * Wn + col;

    const int dy  = row - y;            // window: |dy| <= 12
    const int dxl = col - x;            // leftmost dx of this quad

    v4f v = (v4f)0.0f;
    const bool hit = (dy >= -12) & (dy <= 12) & (dxl <= 12) & (dxl + 3 >= -12);

    if (hit) {
        const float w0 = sw[0], w1 = sw[1], w2 = sw[2];
        // Per-scale "kernel fully inside image" masks (pads 3, 6, 12).
        const bool fit0 = (x >= 3)  & (y >= 3)  & (x + 3  < Wn) & (y + 3  < Hn);
        const bool fit1 = (x >= 6)  & (y >= 6)  & (x + 6  < Wn) & (y + 6  < Hn);
        const bool fit2 = (x >= 12) & (y >= 12) & (x + 12 < Wn) & (y + 12 < Hn);

        const int   ady  = (dy < 0) ? -dy : dy;          // <= 12 here
        const float fdy2 = (float)(dy * dy);

#pragma unroll
        for (int j = 0; j < 4; ++j) {
            const int dx  = dxl + j;
            const int adx = (dx < 0) ? -dx : dx;
            const float d2 = fdy2 + (float)(dx * dx);
            float val = 0.0f;  // reference: max(0, max over fitting scales)
            if (fit2 && adx <= 12)              // sigma=4: 1/(2*16) = 0.03125
                val = fmaxf(val, w2 * expf(-d2 * 0.03125f));
            if (fit1 && adx <= 6 && ady <= 6)   // sigma=2: 1/(2*4)  = 0.125
                val = fmaxf(val, w1 * expf(-d2 * 0.125f));
            if (fit0 && adx <= 3 && ady <= 3)   // sigma=1: 1/2      = 0.5
                val = fmaxf(val, w0 * expf(-d2 * 0.5f));
            v[j] = val;
        }
    }

    // Write-once 126MB stream: non-temporal b128 store (CDNA5 TH_STORE_NT).
    __builtin_nontemporal_store(v, (v4f*)p);
}

extern "C" void kernel_launch(void* const* d_in, const int* in_sizes, int n_in,
                              void* d_out, int out_size, void* d_ws, size_t ws_size,
                              hipStream_t stream) {
    // d_in[0] = image_tensor (float32) -- only defines shapes, never read.
    const int*   kp  = (const int*)d_in[1];    // keypoints [B,K,2] int32
    const float* sw  = (const float*)d_in[2];  // scale_weights [3] float32
    float*       out = (float*)d_out;          // [B,K,H,W] float32

    (void)in_sizes; (void)n_in; (void)out_size; (void)d_ws; (void)ws_size;

    const int grid = Bn * Kn * BLOCKS_PER_CHAN;  // 30720 blocks
    MultiScaleHeatmapGenerator_57458072486269_kernel<<<grid, 256, 0, stream>>>(kp, sw, out);
}